// TransformRCNNOutputs_86388972191788
// MI455X (gfx1250) — compile-verified
//
#include <hip/hip_runtime.h>
#include <math.h>

#define BATCH 8
#define NPROP 1024
#define NCLS  80
#define NCLS1 81
#define KPAD  84
#define REGW  (NCLS * 4)
#define NC    (NPROP * NCLS)
#define SCORE_THR 0.05f
#define IOU_THR   0.5f
#define MAXDET    100
#define MAX_RATIO 4.135166556742356f

typedef float v2f __attribute__((ext_vector_type(2)));
typedef float v8f __attribute__((ext_vector_type(8)));
typedef unsigned int u32;
typedef unsigned long long u64;

// ---------------------------------------------------------------------------
// Kernel 1: softmax over 81 classes for 8192 rows, 128 rows per block.
// Tile staged in LDS via coalesced float4 loads; exp row-sums computed with a
// fully unrolled, branch-free V_WMMA_F32_16X16X4_F32 chain (ones B matrix).
// Scores written transposed: scoresT[B][C][N].
// ---------------------------------------------------------------------------
__global__ __launch_bounds__(256) void softmax_wmma_kernel(
    const float* __restrict__ cls, float* __restrict__ scoresT)
{
    __shared__ float L[128 * KPAD];   // logit/exp tile, K padded 81 -> 84
    __shared__ float sumL[8][16];

    const int tid  = threadIdx.x;
    const int wave = tid >> 5;
    const int lane = tid & 31;
    const int lr   = lane & 15;
    const int half = lane >> 4;
    const int rowBase = blockIdx.x * 128;        // global row of LDS row 0
    const int myRow   = wave * 16 + lr;          // LDS row this lane reduces
    const int rowOff  = myRow * KPAD;

    // Phase A: coalesced float4 global loads -> padded LDS tile
    const float4* g4 = (const float4*)(cls + (size_t)rowBase * NCLS1);
    for (int q = tid; q < (128 * NCLS1) / 4; q += 256) {
        float4 v = g4[q];
        int e = q * 4;
        #pragma unroll
        for (int i = 0; i < 4; ++i) {
            int ei = e + i;
            int r  = ei / NCLS1;
            int k  = ei - r * NCLS1;
            L[r * KPAD + k] = ((const float*)&v)[i];
        }
    }
    for (int t = tid; t < 128 * (KPAD - NCLS1); t += 256) {
        int r = t / (KPAD - NCLS1);
        int k = NCLS1 + (t - r * (KPAD - NCLS1));
        L[r * KPAD + k] = 0.0f;
    }
    if (tid == 0 && (int)blockIdx.x + 1 < (int)gridDim.x)
        __builtin_prefetch(cls + (size_t)(rowBase + 128) * NCLS1, 0, 1);
    __syncthreads();

    // Phase B: row max (both wave-halves reduce the same row, wave-lockstep)
    float maxv = -3.4e38f;
    for (int k = 0; k < NCLS1; ++k) maxv = fmaxf(maxv, L[rowOff + k]);

    // Phase C: exp in place; the two halves split the K range
    {
        int kLo = half ? 41 : 0;
        int kHi = half ? NCLS1 : 41;
        for (int k = kLo; k < kHi; ++k)
            L[rowOff + k] = __expf(L[rowOff + k] - maxv);
    }
    __builtin_amdgcn_wave_barrier();

    // Phase D: branch-free unrolled WMMA row-sum.
    // A (16x4 f32): lanes 0-15 -> K=kb,kb+1 ; lanes 16-31 -> K=kb+2,kb+3.
    v8f acc = {0.f, 0.f, 0.f, 0.f, 0.f, 0.f, 0.f, 0.f};
    v2f bmat = {1.0f, 1.0f};             // all-ones B: D[m][n] = row-sum(A)
    #pragma unroll
    for (int kb = 0; kb < KPAD; kb += 4) {
        v2f a = { L[rowOff + kb + 2 * half], L[rowOff + kb + 2 * half + 1] };
        acc = __builtin_amdgcn_wmma_f32_16x16x4_f32(
            false, a, false, bmat, (short)0, acc, false, false);
    }
    __builtin_amdgcn_wave_barrier();

    // D layout: VGPR i @ lanes 0-15 -> M=i ; lanes 16-31 -> M=i+8
    if (lane == 0) {
        #pragma unroll
        for (int i = 0; i < 8; ++i) sumL[wave][i] = acc[i];
    }
    if (lane == 16) {
        #pragma unroll
        for (int i = 0; i < 8; ++i) sumL[wave][8 + i] = acc[i];
    }
    __builtin_amdgcn_wave_barrier();

    // Phase E: transposed score writes, 16-row bursts per class
    const int gRow0 = rowBase + wave * 16;
    const int bIdx  = gRow0 / NPROP;
    const int n0    = gRow0 - bIdx * NPROP;
    for (int t = lane; t < 16 * NCLS; t += 32) {
        int m = t & 15, c = t >> 4;
        scoresT[((size_t)(bIdx * NCLS + c)) * NPROP + n0 + m] =
            L[(wave * 16 + m) * KPAD + c] / sumL[wave][m];
    }
}

// ---------------------------------------------------------------------------
// Kernel 2: per (batch, class): bitonic sort of 1024 scores (stable
// tie-break), decode boxes in sorted order, greedy NMS on the >thr prefix.
// All per-(b,c) rows are contiguous (transposed layouts) -> coalesced.
// ---------------------------------------------------------------------------
__device__ __forceinline__ u32 ordF(float f) {
    u32 u = __float_as_uint(f);
    return (u & 0x80000000u) ? ~u : (u | 0x80000000u);
}

__global__ __launch_bounds__(256) void sort_nms_kernel(
    const float* __restrict__ scoresT, const float* __restrict__ reg,
    const float* __restrict__ props, const int* __restrict__ ph,
    const int* __restrict__ pw, float* __restrict__ sortedBoxesT,
    float* __restrict__ finalScoresT)
{
    __shared__ u64  keys[NPROP];
    __shared__ float sc[NPROP];
    __shared__ float bx1[NPROP], by1[NPROP], bx2[NPROP], by2[NPROP];
    __shared__ u32  kept[32];
    __shared__ int  cntS;

    const int tid = threadIdx.x;
    const int b = blockIdx.x / NCLS;
    const int c = blockIdx.x % NCLS;
    const float W = (float)pw[0];
    const float H = (float)ph[0];
    const float* srow = scoresT + ((size_t)(b * NCLS + c)) * NPROP;

    // keys: descending score, ascending index on ties (stable argsort(-s))
    for (int j = tid; j < NPROP; j += 256) {
        float s = srow[j];
        keys[j] = ((u64)(~ordF(s)) << 32) | (u32)j;
        __builtin_prefetch(reg + ((size_t)(b * NPROP + j)) * REGW + c * 4, 0, 0);
    }
    if (tid == 0) cntS = 0;
    __syncthreads();

    // bitonic sort ascending on 64-bit keys
    for (int k = 2; k <= NPROP; k <<= 1) {
        for (int j = k >> 1; j > 0; j >>= 1) {
            for (int i = tid; i < NPROP; i += 256) {
                int ixj = i ^ j;
                if (ixj > i) {
                    bool up = ((i & k) == 0);
                    u64 a = keys[i], bb = keys[ixj];
                    bool sw = up ? (a > bb) : (a < bb);
                    if (sw) { keys[i] = bb; keys[ixj] = a; }
                }
            }
            __syncthreads();
        }
    }

    // decode boxes in sorted order; stash in LDS + global (for top-k gather)
    int localCnt = 0;
    for (int j = tid; j < NPROP; j += 256) {
        int n = (int)(u32)keys[j];
        float s = srow[n];
        sc[j] = s;
        if (s > SCORE_THR) localCnt++;
        const float* pb = props + ((size_t)(b * NPROP + n)) * 4;
        const float* dd = reg + ((size_t)(b * NPROP + n)) * REGW + c * 4;
        float dx = dd[0] * 0.1f, dy = dd[1] * 0.1f;
        float dw = dd[2] * 0.2f, dh = dd[3] * 0.2f;
        dw = fminf(fmaxf(dw, -MAX_RATIO), MAX_RATIO);
        dh = fminf(fmaxf(dh, -MAX_RATIO), MAX_RATIO);
        float px1 = pb[0], py1 = pb[1], px2 = pb[2], py2 = pb[3];
        float pxc = (px1 + px2) * 0.5f, pyc = (py1 + py2) * 0.5f;
        float pwd = px2 - px1, pht = py2 - py1;
        float gx = pxc + pwd * dx, gy = pyc + pht * dy;
        float gw = pwd * __expf(dw), gh = pht * __expf(dh);
        float x1 = fminf(fmaxf(gx - gw * 0.5f, 0.f), W);
        float y1 = fminf(fmaxf(gy - gh * 0.5f, 0.f), H);
        float x2 = fminf(fmaxf(gx + gw * 0.5f, 0.f), W);
        float y2 = fminf(fmaxf(gy + gh * 0.5f, 0.f), H);
        bx1[j] = x1; by1[j] = y1; bx2[j] = x2; by2[j] = y2;
        float4 bo = make_float4(x1, y1, x2, y2);
        *(float4*)(sortedBoxesT + (((size_t)(b * NCLS + c)) * NPROP + j) * 4) = bo;
    }
    if (localCnt) atomicAdd(&cntS, localCnt);
    __syncthreads();

    const int M = cntS;  // boxes above SCORE_THR form the sorted prefix [0, M)
    if (tid < 32) {
        int lo = tid * 32;
        u32 w;
        if (M >= lo + 32)      w = 0xFFFFFFFFu;
        else if (M <= lo)      w = 0u;
        else                   w = (1u << (M - lo)) - 1u;
        kept[tid] = w;
    }
    __syncthreads();

    // greedy NMS: only kept i suppress later j (exact reference scan)
    for (int i = 0; i < M; ++i) {
        bool alive = (kept[i >> 5] >> (i & 31)) & 1u;  // uniform branch
        if (alive) {
            float cx1 = bx1[i], cy1 = by1[i], cx2 = bx2[i], cy2 = by2[i];
            float cai = fmaxf(cx2 - cx1, 0.f) * fmaxf(cy2 - cy1, 0.f);
            for (int j = i + 1 + tid; j < M; j += 256) {
                float ix1 = fmaxf(cx1, bx1[j]);
                float iy1 = fmaxf(cy1, by1[j]);
                float ix2 = fminf(cx2, bx2[j]);
                float iy2 = fminf(cy2, by2[j]);
                float inter = fmaxf(ix2 - ix1, 0.f) * fmaxf(iy2 - iy1, 0.f);
                float aj = fmaxf(bx2[j] - bx1[j], 0.f) * fmaxf(by2[j] - by1[j], 0.f);
                float iou = inter / (cai + aj - inter + 1e-6f);
                if (iou > IOU_THR) atomicAnd(&kept[j >> 5], ~(1u << (j & 31)));
            }
        }
        __syncthreads();
    }

    for (int j = tid; j < NPROP; j += 256) {
        bool k = (j < M) && ((kept[j >> 5] >> (j & 31)) & 1u);
        finalScoresT[((size_t)(b * NCLS + c)) * NPROP + j] = k ? sc[j] : 0.0f;
    }
}

// ---------------------------------------------------------------------------
// Kernel 3: per batch, top-100 over the transposed score plane. Tie-breaking
// uses the reference flat index f = n*C + c (value desc, f asc), matching
// jax.lax.top_k on the [N*C] layout, while storage index t = c*N + n.
// ---------------------------------------------------------------------------
__global__ __launch_bounds__(1024) void topk_kernel(
    const float* __restrict__ finalScoresT, const float* __restrict__ sortedBoxesT,
    float* __restrict__ cp, float* __restrict__ det, int* __restrict__ labels)
{
    __shared__ float rv[1024];
    __shared__ int   rf[1024];
    const int tid = threadIdx.x;
    const int b = blockIdx.x;

    for (int t = tid; t < NC; t += 1024)
        cp[(size_t)b * NC + t] = finalScoresT[(size_t)b * NC + t];
    __syncthreads();

    for (int r = 0; r < MAXDET; ++r) {
        float bv = -1.0f; int bf = 0x7FFFFFFF;
        for (int t = tid; t < NC; t += 1024) {
            float v = cp[(size_t)b * NC + t];
            int f = ((t & (NPROP - 1)) * NCLS) + (t >> 10);  // n*C + c
            if (v > bv || (v == bv && f < bf)) { bv = v; bf = f; }
        }
        rv[tid] = bv; rf[tid] = bf;
        __syncthreads();
        for (int s = 512; s > 0; s >>= 1) {
            if (tid < s) {
                float ov = rv[tid + s]; int of = rf[tid + s];
                if (ov > rv[tid] || (ov == rv[tid] && of < rf[tid])) {
                    rv[tid] = ov; rf[tid] = of;
                }
            }
            __syncthreads();
        }
        if (tid == 0) {
            int f = rf[0];
            float val = rv[0];
            int c = f % NCLS, n = f / NCLS;
            int t = c * NPROP + n;                 // transposed storage index
            const float* bo = sortedBoxesT + ((size_t)b * NC + t) * 4;
            float* o = det + ((size_t)(b * MAXDET + r)) * 5;
            o[0] = bo[0]; o[1] = bo[1]; o[2] = bo[2]; o[3] = bo[3]; o[4] = val;
            labels[b * MAXDET + r] = c;
            cp[(size_t)b * NC + t] = -INFINITY;
        }
        __syncthreads();
    }
}

// ---------------------------------------------------------------------------
extern "C" void kernel_launch(void* const* d_in, const int* in_sizes, int n_in,
                              void* d_out, int out_size, void* d_ws, size_t ws_size,
                              hipStream_t stream)
{
    (void)in_sizes; (void)n_in; (void)out_size; (void)ws_size;

    const float* cls   = (const float*)d_in[0];  // [B,N,81]
    const float* reg   = (const float*)d_in[1];  // [B,N,320]
    const float* props = (const float*)d_in[2];  // [B,N,4]
    const int*   ph    = (const int*)d_in[3];    // img_h
    const int*   pw    = (const int*)d_in[4];    // img_w

    float* ws           = (float*)d_ws;
    float* scoresT      = ws;                                    // B*C*N
    float* sortedBoxesT = scoresT + (size_t)BATCH * NC;          // B*C*N*4
    float* finalScoresT = sortedBoxesT + (size_t)BATCH * NC * 4; // B*C*N
    float* cp           = finalScoresT + (size_t)BATCH * NC;     // B*C*N

    float* det    = (float*)d_out;                               // [B,100,5] f32
    int*   labels = (int*)((float*)d_out + BATCH * MAXDET * 5);  // [B,100] i32

    softmax_wmma_kernel<<<(BATCH * NPROP) / 128, 256, 0, stream>>>(cls, scoresT);
    sort_nms_kernel<<<BATCH * NCLS, 256, 0, stream>>>(
        scoresT, reg, props, ph, pw, sortedBoxesT, finalScoresT);
    topk_kernel<<<BATCH, 1024, 0, stream>>>(
        finalScoresT, sortedBoxesT, cp, det, labels);
}